// GAT_STUDENT_23716809409155
// MI455X (gfx1250) — compile-verified
//
#include <hip/hip_runtime.h>
#include <hip/hip_bf16.h>

// GAT forward, MI455X (gfx1250, wave32).
//   inputs: x[4096,4096] f32, adj[4096,4096] i32, W[4,4096,8] f32,
//           a[4,16] f32, w_lin[1,4096] f32, b_lin[1] f32
//   outputs (concat): out[1,8] f32, node_embeddings[4096,8] f32
//
// Pipeline:
//   K1: Wh[4096,32] = x @ Wcat            (v_wmma_f32_16x16x4_f32, heads packed)
//   K2: s_src/s_dst scores
//   K3: per-head global softmax shift M_h = lrelu(max s_src + max s_dst)
//   K4: fused mask+softmax+aggregate, WMMA per head, ones-column = denominator
//   K5: combine j-partials, head-mean, log_softmax
//   K6: final linear reduction
//
// HBM ~ x(64MB) + adj(64MB) => ~5.5us floor at 23.3 TB/s. Wh/s_dst L2-resident.

typedef __attribute__((ext_vector_type(2))) float v2f;
typedef __attribute__((ext_vector_type(8))) float v8f;

#define GAT_N      4096
#define GAT_NC     8
#define GAT_H      4
#define GAT_COLS   32
#define GAT_ALPHA  0.2f
#define GAT_NEGINF (-9.0e15f)

#define CHJ        128            // j-chunk per LDS stage
#define WH_STRIDE  36             // padded LDS row stride for Wh tile (16B-aligned, conflict-free)
#define ADJ_STRIDE 17             // padded LDS col stride for adj tile (conflict-free transpose)

// ---------------------------------------------------------------------------
// K1: Wh = x @ Wcat, Wcat(f, c) = W[(c>>3)*4096*8 + f*8 + (c&7)]
// 64 blocks x 128 threads (4 waves); wave -> 16-row M tile, two 16x16 N tiles.
// ---------------------------------------------------------------------------
__global__ void gat_gemm_wh_kernel(const float* __restrict__ x,
                                   const float* __restrict__ W,
                                   float* __restrict__ Wh) {
  const int lane = threadIdx.x & 31;
  const int wave = threadIdx.x >> 5;
  const int m0   = blockIdx.x * 64 + wave * 16;
  const int half = lane >> 4;
  const int l    = lane & 15;

  v8f c0 = {};
  v8f c1 = {};

  const float* xrow = x + (size_t)(m0 + l) * GAT_N;

  const int col0 = l;
  const int col1 = 16 + l;
  const size_t b0base = (size_t)(col0 >> 3) * (GAT_N * GAT_NC) + (col0 & 7);
  const size_t b1base = (size_t)(col1 >> 3) * (GAT_N * GAT_NC) + (col1 & 7);

  for (int k = 0; k < GAT_N; k += 4) {
    const int kk = k + half * 2;
    v2f af;
    af.x = xrow[kk];
    af.y = xrow[kk + 1];
    v2f b0f, b1f;
    b0f.x = W[b0base + (size_t)kk * 8];
    b0f.y = W[b0base + (size_t)(kk + 1) * 8];
    b1f.x = W[b1base + (size_t)kk * 8];
    b1f.y = W[b1base + (size_t)(kk + 1) * 8];

    c0 = __builtin_amdgcn_wmma_f32_16x16x4_f32(false, af, false, b0f,
                                               (short)0, c0, false, false);
    c1 = __builtin_amdgcn_wmma_f32_16x16x4_f32(false, af, false, b1f,
                                               (short)0, c1, false, false);
  }

  for (int r = 0; r < 8; ++r) {
    const int row = m0 + r + half * 8;
    Wh[(size_t)row * GAT_COLS + col0] = c0[r];
    Wh[(size_t)row * GAT_COLS + col1] = c1[r];
  }
}

// ---------------------------------------------------------------------------
// K2: s_src[h*N+n] = Wh[n,h*8:] . a[h,:8] ; s_dst with a[h,8:]
// ---------------------------------------------------------------------------
__global__ void gat_scores_kernel(const float* __restrict__ Wh,
                                  const float* __restrict__ a,
                                  float* __restrict__ s_src,
                                  float* __restrict__ s_dst) {
  const int n = blockIdx.x * blockDim.x + threadIdx.x;
  if (n >= GAT_N) return;
  const float* whr = Wh + (size_t)n * GAT_COLS;
  for (int h = 0; h < GAT_H; ++h) {
    float ss = 0.f, sd = 0.f;
    for (int o = 0; o < GAT_NC; ++o) {
      const float w = whr[h * GAT_NC + o];
      ss += w * a[h * 16 + o];
      sd += w * a[h * 16 + 8 + o];
    }
    s_src[h * GAT_N + n] = ss;
    s_dst[h * GAT_N + n] = sd;
  }
}

// ---------------------------------------------------------------------------
// K3: gmax[h] = lrelu(max_i s_src[h,i] + max_j s_dst[h,j])
// Valid global softmax shift: >= every masked row max, exp(e-M) <= 1.
// ---------------------------------------------------------------------------
__global__ void gat_gmax_kernel(const float* __restrict__ s_src,
                                const float* __restrict__ s_dst,
                                float* __restrict__ gmax) {
  __shared__ float red[256];
  const int tid = threadIdx.x;
  for (int h = 0; h < GAT_H; ++h) {
    float m1 = -3.4e38f, m2 = -3.4e38f;
    for (int n = tid; n < GAT_N; n += 256) {
      m1 = fmaxf(m1, s_src[h * GAT_N + n]);
      m2 = fmaxf(m2, s_dst[h * GAT_N + n]);
    }
    red[tid] = m1;
    __syncthreads();
    for (int s = 128; s > 0; s >>= 1) {
      if (tid < s) red[tid] = fmaxf(red[tid], red[tid + s]);
      __syncthreads();
    }
    const float M1 = red[0];
    __syncthreads();
    red[tid] = m2;
    __syncthreads();
    for (int s = 128; s > 0; s >>= 1) {
      if (tid < s) red[tid] = fmaxf(red[tid], red[tid + s]);
      __syncthreads();
    }
    if (tid == 0) {
      const float v = M1 + red[0];
      gmax[h] = (v >= 0.f) ? v : GAT_ALPHA * v;
    }
    __syncthreads();
  }
}

// ---------------------------------------------------------------------------
// K4: fused attention. grid (256 i-tiles, 2 j-halves) x 128 threads (4 waves).
// Wave h: A = exp(masked lrelu scores) in native WMMA A layout,
//         B = Wh[j, h*8+c] for c<8, ones at c==8 (denominator), 0 above.
// D tile: rows = 16 dst nodes, cols 0..7 = unnorm h_prime, col 8 = sum(p).
// ---------------------------------------------------------------------------
__global__ void gat_attn_wmma_kernel(const int* __restrict__ adj,
                                     const float* __restrict__ Wh,
                                     const float* __restrict__ s_src,
                                     const float* __restrict__ s_dst,
                                     const float* __restrict__ gmax,
                                     float* __restrict__ part) {
  const int it   = blockIdx.x;
  const int jh   = blockIdx.y;
  const int i0   = it * 16;
  const int tid  = threadIdx.x;     // 0..127
  const int lane = tid & 31;
  const int h    = tid >> 5;        // wave == head
  const int half = lane >> 4;       // A/B K-half (K 0,1 vs 2,3)
  const int r    = lane & 15;       // A: M row; B: N col

  __shared__ __align__(16) float s_wh[CHJ * WH_STRIDE];
  __shared__ int   s_adj[CHJ * ADJ_STRIDE];
  __shared__ __align__(16) float s_sd[GAT_H * CHJ];

  const float srow = s_src[h * GAT_N + i0 + r];
  const float gm   = gmax[h];

  v8f acc = {};

  const int jbase = jh * (GAT_N / 2);
  for (int jc = jbase; jc < jbase + GAT_N / 2; jc += CHJ) {
    // ---- cooperative stage to LDS ----
    // Wh tile: CHJ x 32 floats, padded stride 36 (float4 in/out, bank-clean)
    for (int k = 0; k < 8; ++k) {
      const int flat4 = k * 128 + tid;           // 1024 float4 slots
      const int j = flat4 >> 3;
      const int c = (flat4 & 7) << 2;
      const float4 v = *(const float4*)&Wh[(size_t)(jc + j) * GAT_COLS + c];
      *(float4*)&s_wh[j * WH_STRIDE + c] = v;
    }
    // adj tile: 16 rows x CHJ, stored transposed s_adj[col*17 + row]
    for (int k = 0; k < 4; ++k) {
      const int flat4 = k * 128 + tid;           // 512 int4 slots
      const int row = flat4 >> 5;
      const int c4  = (flat4 & 31) << 2;
      const int4 v = *(const int4*)&adj[(size_t)(i0 + row) * GAT_N + jc + c4];
      s_adj[(c4 + 0) * ADJ_STRIDE + row] = v.x;
      s_adj[(c4 + 1) * ADJ_STRIDE + row] = v.y;
      s_adj[(c4 + 2) * ADJ_STRIDE + row] = v.z;
      s_adj[(c4 + 3) * ADJ_STRIDE + row] = v.w;
    }
    // s_dst slice: 4 heads x CHJ
    {
      const int hh = tid >> 5;
      const int c4 = (tid & 31) << 2;
      const float4 v = *(const float4*)&s_dst[hh * GAT_N + jc + c4];
      *(float4*)&s_sd[hh * CHJ + c4] = v;
    }
    __syncthreads();

    // prefetch next adj chunk while computing (global_prefetch_b8)
    if (jc + CHJ < jbase + GAT_N / 2)
      __builtin_prefetch(&adj[(size_t)(i0 + (tid >> 3)) * GAT_N + jc + CHJ +
                              ((tid & 7) << 4)], 0, 0);

    // ---- exp + WMMA accumulate ----
    const float* __restrict__ sdh = s_sd + h * CHJ;
    for (int jj = 0; jj < CHJ; jj += 4) {
      const int kk = jj + half * 2;
      const int a0 = s_adj[kk * ADJ_STRIDE + r];
      const int a1 = s_adj[(kk + 1) * ADJ_STRIDE + r];
      float v0 = srow + sdh[kk];
      float v1 = srow + sdh[kk + 1];
      v0 = (v0 >= 0.f) ? v0 : GAT_ALPHA * v0;
      v1 = (v1 >= 0.f) ? v1 : GAT_ALPHA * v1;
      const float m0 = (a0 > 0) ? v0 : GAT_NEGINF;
      const float m1 = (a1 > 0) ? v1 : GAT_NEGINF;
      v2f af;
      af.x = __expf(m0 - gm);                    // masked -> exp(-9e15) = 0
      af.y = __expf(m1 - gm);
      v2f bf;
      bf.x = (r < 8) ? s_wh[kk * WH_STRIDE + h * 8 + r]
                     : ((r == 8) ? 1.0f : 0.0f);
      bf.y = (r < 8) ? s_wh[(kk + 1) * WH_STRIDE + h * 8 + r]
                     : ((r == 8) ? 1.0f : 0.0f);
      acc = __builtin_amdgcn_wmma_f32_16x16x4_f32(false, af, false, bf,
                                                  (short)0, acc, false, false);
    }
    __syncthreads();
  }

  // D layout: acc[v] on lane -> (M = v + half*8, N = r); store partial tile
  float* pt = part + (((size_t)jh * 256 + it) * GAT_H + h) * 256;
  for (int v = 0; v < 8; ++v)
    pt[(v + half * 8) * 16 + r] = acc[v];
}

// ---------------------------------------------------------------------------
// K5: combine j-halves (linear thanks to global shift), normalize, head-mean,
//     log_softmax.  One thread per node.
// ---------------------------------------------------------------------------
__global__ void gat_combine_kernel(const float* __restrict__ part,
                                   float* __restrict__ node_emb,
                                   float* __restrict__ ls) {
  const int n = blockIdx.x * blockDim.x + threadIdx.x;
  if (n >= GAT_N) return;
  const int it = n >> 4;
  const int m  = n & 15;

  float emb[GAT_NC];
  for (int o = 0; o < GAT_NC; ++o) emb[o] = 0.f;
  for (int h = 0; h < GAT_H; ++h) {
    const float* p0 = part + (((size_t)0 * 256 + it) * GAT_H + h) * 256 + m * 16;
    const float* p1 = part + (((size_t)1 * 256 + it) * GAT_H + h) * 256 + m * 16;
    const float den = p0[8] + p1[8];
    const float inv = 1.0f / den;
    for (int o = 0; o < GAT_NC; ++o) emb[o] += (p0[o] + p1[o]) * inv;
  }
  float mx = -3.4e38f;
  for (int o = 0; o < GAT_NC; ++o) {
    emb[o] *= 0.25f;
    node_emb[(size_t)n * GAT_NC + o] = emb[o];
    mx = fmaxf(mx, emb[o]);
  }
  float s = 0.f;
  for (int o = 0; o < GAT_NC; ++o) s += expf(emb[o] - mx);
  const float lse = mx + logf(s);
  for (int o = 0; o < GAT_NC; ++o)
    ls[(size_t)n * GAT_NC + o] = emb[o] - lse;
}

// ---------------------------------------------------------------------------
// K6: out[c] = b_lin + sum_n ls[n,c] * w_lin[n]
// ---------------------------------------------------------------------------
__global__ void gat_final_linear_kernel(const float* __restrict__ ls,
                                        const float* __restrict__ w_lin,
                                        const float* __restrict__ b_lin,
                                        float* __restrict__ out) {
  __shared__ float buf[256 * GAT_NC];
  const int tid = threadIdx.x;
  float p[GAT_NC] = {};
  for (int n = tid; n < GAT_N; n += 256) {
    const float w = w_lin[n];
    for (int o = 0; o < GAT_NC; ++o) p[o] += ls[(size_t)n * GAT_NC + o] * w;
  }
  for (int o = 0; o < GAT_NC; ++o) buf[tid * GAT_NC + o] = p[o];
  __syncthreads();
  if (tid < GAT_NC) {
    float s = b_lin[0];
    for (int t = 0; t < 256; ++t) s += buf[t * GAT_NC + tid];
    out[tid] = s;
  }
}

// ---------------------------------------------------------------------------
extern "C" void kernel_launch(void* const* d_in, const int* in_sizes, int n_in,
                              void* d_out, int out_size, void* d_ws, size_t ws_size,
                              hipStream_t stream) {
  const float* x     = (const float*)d_in[0];
  const int*   adj   = (const int*)d_in[1];
  const float* W     = (const float*)d_in[2];
  const float* a     = (const float*)d_in[3];
  const float* w_lin = (const float*)d_in[4];
  const float* b_lin = (const float*)d_in[5];

  float* out      = (float*)d_out;
  float* node_emb = (float*)d_out + GAT_NC;

  float* ws    = (float*)d_ws;
  float* Wh    = ws;                         // 131072 f32
  float* s_src = Wh + GAT_N * GAT_COLS;      // 16384
  float* s_dst = s_src + GAT_H * GAT_N;      // 16384
  float* ls    = s_dst + GAT_H * GAT_N;      // 32768
  float* gmax  = ls + GAT_N * GAT_NC;        // 4 (+pad)
  float* part  = gmax + 32;                  // 2*256*4*256 = 524288 f32

  gat_gemm_wh_kernel<<<GAT_N / 64, 128, 0, stream>>>(x, W, Wh);
  gat_scores_kernel<<<GAT_N / 256, 256, 0, stream>>>(Wh, a, s_src, s_dst);
  gat_gmax_kernel<<<1, 256, 0, stream>>>(s_src, s_dst, gmax);
  gat_attn_wmma_kernel<<<dim3(GAT_N / 16, 2), 128, 0, stream>>>(
      adj, Wh, s_src, s_dst, gmax, part);
  gat_combine_kernel<<<GAT_N / 256, 256, 0, stream>>>(part, node_emb, ls);
  gat_final_linear_kernel<<<1, 256, 0, stream>>>(ls, w_lin, b_lin, out);
}